// GraphConv_72060961292958
// MI455X (gfx1250) — compile-verified
//
#include <hip/hip_runtime.h>

// Problem shape (from reference setup_inputs)
#define N_NODES 100000
#define E_DIM   128     // embedding dim (K of the GEMM)
#define M_NBR   16      // neighbors per node
#define FOUT    256     // output features (N of the GEMM)
// B (query count) taken from in_sizes at launch.

typedef float v2f __attribute__((ext_vector_type(2)));
typedef float v8f __attribute__((ext_vector_type(8)));

// ---------------------------------------------------------------------------
// hop1: h1[n][:] = mean_j emb[neighbors[n][j]][:]   for all n
// One wave32 per node; lane owns float4 slice -> 512B coalesced row loads.
// ---------------------------------------------------------------------------
__global__ __launch_bounds__(256) void hop1_kernel(
    const float* __restrict__ emb,
    const int*   __restrict__ neighbors,
    float*       __restrict__ h1,
    int n_nodes)
{
    const int gwave = (int)((blockIdx.x * 256u + threadIdx.x) >> 5);
    const int lane  = threadIdx.x & 31;
    if (gwave >= n_nodes) return;

    const int* nb = neighbors + (size_t)gwave * M_NBR;
    float4 acc = {0.f, 0.f, 0.f, 0.f};
#pragma unroll
    for (int j = 0; j < M_NBR; ++j) {
        const int nbr = nb[j];  // wave-uniform address; HW coalesces
        const float4 v = ((const float4*)(emb + (size_t)nbr * E_DIM))[lane];
        acc.x += v.x; acc.y += v.y; acc.z += v.z; acc.w += v.w;
    }
    const float s = 1.0f / (float)M_NBR;
    float4 o = {acc.x * s, acc.y * s, acc.z * s, acc.w * s};
    ((float4*)(h1 + (size_t)gwave * E_DIM))[lane] = o;
}

// ---------------------------------------------------------------------------
// hop2 at query nodes only, fused with the self-embedding add:
//   X[b][:] = mean_j h1[neighbors[inputs[b]][j]][:] + emb[inputs[b]][:]
// ---------------------------------------------------------------------------
__global__ __launch_bounds__(256) void hop2q_kernel(
    const float* __restrict__ h1,
    const float* __restrict__ emb,
    const int*   __restrict__ neighbors,
    const int*   __restrict__ inputs,
    float*       __restrict__ X,
    int bq)
{
    const int gwave = (int)((blockIdx.x * 256u + threadIdx.x) >> 5);
    const int lane  = threadIdx.x & 31;
    if (gwave >= bq) return;

    const int node = inputs[gwave];
    const int* nb  = neighbors + (size_t)node * M_NBR;

    float4 acc = {0.f, 0.f, 0.f, 0.f};
#pragma unroll
    for (int j = 0; j < M_NBR; ++j) {
        const int nbr = nb[j];
        const float4 v = ((const float4*)(h1 + (size_t)nbr * E_DIM))[lane];
        acc.x += v.x; acc.y += v.y; acc.z += v.z; acc.w += v.w;
    }
    const float s = 1.0f / (float)M_NBR;
    const float4 se = ((const float4*)(emb + (size_t)node * E_DIM))[lane];
    float4 o = {acc.x * s + se.x, acc.y * s + se.y,
                acc.z * s + se.z, acc.w * s + se.w};
    ((float4*)(X + (size_t)gwave * E_DIM))[lane] = o;
}

// ---------------------------------------------------------------------------
// GEMM: out = relu(X @ weight^T + bias), fp32-exact via V_WMMA_F32_16X16X4_F32
// Block = 256 threads = 8 waves. Block computes a 16-row batch tile x all 256
// output cols; wave w computes col-tiles w*16 and (w+8)*16.
// A fragments from padded LDS (ds_load_b64, conflict-free); B fragments are
// contiguous float2 loads from weight (L2-hot, 128KB total).
// ---------------------------------------------------------------------------
__global__ __launch_bounds__(256) void gemm_relu_kernel(
    const float* __restrict__ X,
    const float* __restrict__ weight,   // [FOUT][E_DIM] row-major
    const float* __restrict__ bias,
    float*       __restrict__ out)      // [B][FOUT]
{
    __shared__ float xs[16][E_DIM + 4];   // +4 pad: row stride 132 floats

    const int tid  = threadIdx.x;
    const int wave = tid >> 5;
    const int lane = tid & 31;
    const int half = lane >> 4;   // 0: lanes 0-15, 1: lanes 16-31
    const int l16  = lane & 15;
    const size_t row0 = (size_t)blockIdx.x * 16;

    // Cooperative, coalesced load of the 16 x 128 X tile into LDS.
    // 16 rows * 32 float4 = 512 float4; 256 threads -> 2 each. No divergence.
#pragma unroll
    for (int i = tid; i < 16 * (E_DIM / 4); i += 256) {
        const int r  = i >> 5;        // E_DIM/4 == 32
        const int c4 = i & 31;
        const float4 v = ((const float4*)(X + (row0 + r) * E_DIM))[c4];
        *(float4*)&xs[r][c4 * 4] = v; // row offset 528B -> 16B aligned
    }
    __syncthreads();

    const int colA = wave * 16;
    const int colB = (wave + 8) * 16;

    v8f acc0 = {0.f,0.f,0.f,0.f,0.f,0.f,0.f,0.f};
    v8f acc1 = {0.f,0.f,0.f,0.f,0.f,0.f,0.f,0.f};

#pragma unroll
    for (int k = 0; k < E_DIM; k += 4) {
        const int ke = k + half * 2;  // lanes 0-15: K=k,k+1; lanes 16-31: K=k+2,k+3
        // A fragment (16x4): row = l16 in both halves
        const float2 a2 = *(const float2*)&xs[l16][ke];
        const v2f a = {a2.x, a2.y};
        // B fragments (4x16): col = colT + l16 in both halves; W^T[k][col] = weight[col][k]
        const float2 w0 = *(const float2*)&weight[(size_t)(colA + l16) * E_DIM + ke];
        const float2 w1 = *(const float2*)&weight[(size_t)(colB + l16) * E_DIM + ke];
        const v2f b0 = {w0.x, w0.y};
        const v2f b1 = {w1.x, w1.y};
        acc0 = __builtin_amdgcn_wmma_f32_16x16x4_f32(false, a, false, b0,
                                                     (short)0, acc0, false, false);
        acc1 = __builtin_amdgcn_wmma_f32_16x16x4_f32(false, a, false, b1,
                                                     (short)0, acc1, false, false);
    }

    // Epilogue: bias + relu + store. D layout: VGPR r -> M = r + 8*half, N = l16.
    const float bA = bias[colA + l16];
    const float bB = bias[colB + l16];
#pragma unroll
    for (int r = 0; r < 8; ++r) {
        const size_t row = row0 + (size_t)(r + half * 8);
        float v0 = acc0[r] + bA; v0 = v0 > 0.f ? v0 : 0.f;
        float v1 = acc1[r] + bB; v1 = v1 > 0.f ? v1 : 0.f;
        out[row * FOUT + colA + l16] = v0;
        out[row * FOUT + colB + l16] = v1;
    }
}

// ---------------------------------------------------------------------------
// Launch
// ---------------------------------------------------------------------------
extern "C" void kernel_launch(void* const* d_in, const int* in_sizes, int n_in,
                              void* d_out, int out_size, void* d_ws, size_t ws_size,
                              hipStream_t stream)
{
    (void)n_in; (void)out_size; (void)ws_size;

    // setup_inputs order: inputs, neighbors, emb, weight, bias
    const int*   inputs    = (const int*)d_in[0];
    const int*   neighbors = (const int*)d_in[1];
    const float* emb       = (const float*)d_in[2];
    const float* weight    = (const float*)d_in[3];
    const float* bias      = (const float*)d_in[4];
    float*       out       = (float*)d_out;

    const int bq      = in_sizes[0];               // 16384
    const int n_nodes = in_sizes[1] / M_NBR;       // 100000

    // Workspace layout: h1 [N*E], X [B*E]  (~59.6 MB total)
    float* h1 = (float*)d_ws;
    float* X  = h1 + (size_t)n_nodes * E_DIM;

    // hop1 over all nodes: one wave per node, 8 waves per 256-thread block
    {
        const int blocks = (n_nodes + 7) / 8;
        hop1_kernel<<<blocks, 256, 0, stream>>>(emb, neighbors, h1, n_nodes);
    }
    // hop2 at query nodes + self-embedding add
    {
        const int blocks = (bq + 7) / 8;
        hop2q_kernel<<<blocks, 256, 0, stream>>>(h1, emb, neighbors, inputs, X, bq);
    }
    // fused GEMM + bias + relu (bq is a multiple of 16)
    {
        const int blocks = bq / 16;
        gemm_relu_kernel<<<blocks, 256, 0, stream>>>(X, weight, bias, out);
    }
}